// DGAModel_73400991088769
// MI455X (gfx1250) — compile-verified
//
#include <hip/hip_runtime.h>

typedef __attribute__((ext_vector_type(2))) float v2f;
typedef __attribute__((ext_vector_type(8))) float v8f;

#define B_  8
#define S_  512
#define D_  1024
#define H_  16
#define DK_ 64

// fp32 matrix core op: D(16x16) = A(16x4) * B(4x16) + C
static __device__ __forceinline__ v8f wmma4(v2f a, v2f b, v8f c) {
  return __builtin_amdgcn_wmma_f32_16x16x4_f32(false, a, false, b, (short)0, c,
                                               false, false);
}

// Async global->LDS copy (16B per lane), tracked by ASYNCcnt.
static __device__ __forceinline__ void async_ld_b128(void* lds, const void* g) {
  const unsigned lo = (unsigned)(unsigned long long)lds;  // LDS offset = low 32b
  asm volatile("global_load_async_to_lds_b128 %0, %1, off"
               :: "v"(lo), "v"(g) : "memory");
}
static __device__ __forceinline__ void wait_async0() {
  asm volatile("s_wait_asynccnt 0" ::: "memory");
}

// ---------------------------------------------------------------------------
// Tiled GEMM with bias: Y[M,N] = X[M,K] @ W[K,N] + bias[N]
// 256 threads (8 waves), 64x64 output tile, fp32 WMMA.
// X tile staged via async-to-LDS; W tile staged transposed (Wt[n][k]).
// ---------------------------------------------------------------------------
__global__ __launch_bounds__(256)
void gemm_bias_kernel(const float* __restrict__ X, const float* __restrict__ W,
                      const float* __restrict__ bias, float* __restrict__ Y,
                      int M, int K, int N) {
  __shared__ float Xs[64 * 36];   // 64 m-rows x 32 k   (stride 36)
  __shared__ float Wt[64 * 36];   // 64 n-rows x 32 k   (transposed, stride 36)

  const int t    = threadIdx.x;
  const int wv   = t >> 5;
  const int lane = t & 31;
  const int lh   = lane >> 4;
  const int l    = lane & 15;
  const int wr   = wv >> 1;     // 0..3 (16-row band)
  const int wc   = wv & 1;      // 0..1 (32-col band)
  const int Mb   = blockIdx.y * 64;
  const int Nb   = blockIdx.x * 64;

  v8f c0 = {}; v8f c1 = {};

  for (int k0 = 0; k0 < K; k0 += 32) {
    // X tile 64x32: async global->LDS, contiguous 16B chunks
    for (int i = t; i < 512; i += 256) {
      const int row = i >> 3, kc = (i & 7) << 2;
      async_ld_b128(Xs + row * 36 + kc,
                    X + (size_t)(Mb + row) * K + k0 + kc);
    }
    // W tile 32x64 -> transposed into Wt[n][k]
    for (int i = t; i < 512; i += 256) {
      const int row = i >> 4, nc = (i & 15) << 2;
      const float4 v = *(const float4*)(W + (size_t)(k0 + row) * N + Nb + nc);
      Wt[(nc + 0) * 36 + row] = v.x;
      Wt[(nc + 1) * 36 + row] = v.y;
      Wt[(nc + 2) * 36 + row] = v.z;
      Wt[(nc + 3) * 36 + row] = v.w;
    }
    wait_async0();
    __syncthreads();
#pragma unroll
    for (int kk = 0; kk < 32; kk += 4) {
      const int ka = kk + lh * 2;
      const v2f a  = *(const v2f*)(Xs + (wr * 16 + l) * 36 + ka);
      const v2f b0 = *(const v2f*)(Wt + (wc * 32 + l) * 36 + ka);
      const v2f b1 = *(const v2f*)(Wt + (wc * 32 + 16 + l) * 36 + ka);
      c0 = wmma4(a, b0, c0);
      c1 = wmma4(a, b1, c1);
    }
    __syncthreads();
  }
#pragma unroll
  for (int r = 0; r < 8; ++r) {
    const int gm  = Mb + wr * 16 + lh * 8 + r;
    const int gn0 = Nb + wc * 32 + l;
    const int gn1 = gn0 + 16;
    Y[(size_t)gm * N + gn0] = c0[r] + bias[gn0];
    Y[(size_t)gm * N + gn1] = c1[r] + bias[gn1];
  }
}

// ---------------------------------------------------------------------------
// Attention: per workgroup = (b, h, 32 q-rows).
// scores = QK^T/8 -> attn_mask -> dependency blend -> softmax -> attn out
// ctx = attn @ V  (written to Ctx in [B*S, H*DK] layout)
// ---------------------------------------------------------------------------
#define QROWS  32
#define QS_STR 68
#define KT_STR 68
#define SB_STR 516

__global__ __launch_bounds__(256)
void attention_kernel(const float* __restrict__ Qb, const float* __restrict__ Kb,
                      const float* __restrict__ Vb,
                      const float* __restrict__ help_vals,
                      const int* __restrict__ help_mask,
                      const unsigned char* __restrict__ attn_mask,
                      float* __restrict__ attnOut, float* __restrict__ Ctx) {
  extern __shared__ float smem[];
  float* Qs = smem;                         // 32 x 68  (q-row x dk)
  float* Kt = Qs + QROWS * QS_STR;          // 64 x 68  (K: n x dk; V: n x k)
  float* Sb = Kt + 64 * KT_STR;             // 32 x 516 (scores / attn)

  const int t    = threadIdx.x;
  const int wv   = t >> 5;
  const int lane = t & 31;
  const int lh   = lane >> 4;
  const int l    = lane & 15;
  const int wr   = wv >> 2;     // 0..1 (16-row band of the 32 q rows)
  const int wc   = wv & 3;      // 0..3 (16-col band of 64)
  const int q0   = blockIdx.x * QROWS;
  const int h    = blockIdx.y;
  const int b    = blockIdx.z;

  // Q tile: 32 rows x 64 dk (async)
  for (int i = t; i < QROWS * 16; i += 256) {
    const int row = i >> 4, dc = (i & 15) << 2;
    async_ld_b128(Qs + row * QS_STR + dc,
                  Qb + (size_t)(b * S_ + q0 + row) * D_ + h * DK_ + dc);
  }
  wait_async0();
  __syncthreads();

  // ---- scores = Q K^T * 0.125 ----
  for (int kt = 0; kt < S_ / 64; ++kt) {
    for (int i = t; i < 64 * 16; i += 256) {
      const int row = i >> 4, dc = (i & 15) << 2;
      async_ld_b128(Kt + row * KT_STR + dc,
                    Kb + (size_t)(b * S_ + kt * 64 + row) * D_ + h * DK_ + dc);
    }
    wait_async0();
    __syncthreads();
    v8f c = {};
#pragma unroll
    for (int kk = 0; kk < DK_; kk += 4) {
      const int ka = kk + lh * 2;
      const v2f a  = *(const v2f*)(Qs + (wr * 16 + l) * QS_STR + ka);
      const v2f bb = *(const v2f*)(Kt + (wc * 16 + l) * KT_STR + ka);
      c = wmma4(a, bb, c);
    }
#pragma unroll
    for (int r = 0; r < 8; ++r)
      Sb[(wr * 16 + lh * 8 + r) * SB_STR + kt * 64 + wc * 16 + l] = c[r] * 0.125f;
    __syncthreads();
  }

  // ---- attn_mask + dependency-tree blend (help streams dominate BW) ----
  for (int i = t; i < QROWS * S_; i += 256) {
    const int row = i >> 9;
    const int col = i & 511;
    const int qr  = q0 + row;
    float sc = Sb[row * SB_STR + col];
    if (attn_mask[((size_t)b * S_ + qr) * S_ + col]) sc = -1e9f;
    const size_t hidx = (((size_t)(b * H_ + h) * S_) + qr) * S_ + col;
    if (i + 256 < QROWS * S_) {
      __builtin_prefetch(help_vals + hidx + 256, 0, 1);
      __builtin_prefetch(help_mask + hidx + 256, 0, 1);
    }
    const float hs = help_mask[hidx] ? help_vals[hidx] : -1e9f;
    if (hs > -1e8f) sc = 0.5f * (hs + sc);
    Sb[row * SB_STR + col] = sc;
  }
  __syncthreads();

  // ---- softmax per row (4 rows per wave), write attn output ----
  for (int j = 0; j < 4; ++j) {
    const int row = wv * 4 + j;
    float* srow = Sb + row * SB_STR;
    float mx = -3.4e38f;
#pragma unroll
    for (int i = 0; i < 16; ++i) mx = fmaxf(mx, srow[lane + 32 * i]);
#pragma unroll
    for (int off = 16; off > 0; off >>= 1) mx = fmaxf(mx, __shfl_xor(mx, off, 32));
    float e[16];
    float sum = 0.f;
#pragma unroll
    for (int i = 0; i < 16; ++i) { e[i] = expf(srow[lane + 32 * i] - mx); sum += e[i]; }
#pragma unroll
    for (int off = 16; off > 0; off >>= 1) sum += __shfl_xor(sum, off, 32);
    const float inv = 1.0f / sum;
    float* arow = attnOut + (((size_t)(b * H_ + h) * S_) + q0 + row) * S_;
#pragma unroll
    for (int i = 0; i < 16; ++i) {
      const float av = e[i] * inv;
      srow[lane + 32 * i] = av;
      arow[lane + 32 * i] = av;
    }
  }
  __syncthreads();

  // ---- ctx = attn @ V ----  (V tile staged transposed: Vt[n][k])
  v8f cc = {};
  for (int vt = 0; vt < S_ / 64; ++vt) {
    for (int i = t; i < 64 * 16; i += 256) {
      const int row = i >> 4, dc = (i & 15) << 2;
      const float4 v = *(const float4*)(Vb + (size_t)(b * S_ + vt * 64 + row) * D_ +
                                        h * DK_ + dc);
      Kt[(dc + 0) * KT_STR + row] = v.x;
      Kt[(dc + 1) * KT_STR + row] = v.y;
      Kt[(dc + 2) * KT_STR + row] = v.z;
      Kt[(dc + 3) * KT_STR + row] = v.w;
    }
    __syncthreads();
#pragma unroll
    for (int kk = 0; kk < 64; kk += 4) {
      const int ka = kk + lh * 2;
      const v2f a  = *(const v2f*)(Sb + (wr * 16 + l) * SB_STR + vt * 64 + ka);
      const v2f bb = *(const v2f*)(Kt + (wc * 16 + l) * KT_STR + ka);
      cc = wmma4(a, bb, cc);
    }
    __syncthreads();
  }
#pragma unroll
  for (int r = 0; r < 8; ++r) {
    const int qr = q0 + wr * 16 + lh * 8 + r;
    Ctx[(size_t)(b * S_ + qr) * D_ + h * DK_ + wc * 16 + l] = cc[r];
  }
}

// ---------------------------------------------------------------------------
// Residual + LayerNorm: y = LN(proj + x) * g + b ; one row per workgroup.
// ---------------------------------------------------------------------------
__global__ __launch_bounds__(256)
void resid_ln_kernel(const float* __restrict__ proj, const float* __restrict__ x,
                     const float* __restrict__ g, const float* __restrict__ bta,
                     float* __restrict__ y) {
  __shared__ float red[8];
  const size_t row = blockIdx.x;
  const int t = threadIdx.x, wv = t >> 5, lane = t & 31;
  float v[4];
#pragma unroll
  for (int i = 0; i < 4; ++i) {
    const int col = t + 256 * i;
    v[i] = proj[row * D_ + col] + x[row * D_ + col];
  }
  float s = v[0] + v[1] + v[2] + v[3];
#pragma unroll
  for (int off = 16; off > 0; off >>= 1) s += __shfl_xor(s, off, 32);
  if (lane == 0) red[wv] = s;
  __syncthreads();
  float tot = 0.f;
#pragma unroll
  for (int i = 0; i < 8; ++i) tot += red[i];
  const float mu = tot * (1.0f / D_);
  __syncthreads();
  float vs = 0.f;
#pragma unroll
  for (int i = 0; i < 4; ++i) { const float d = v[i] - mu; vs += d * d; }
#pragma unroll
  for (int off = 16; off > 0; off >>= 1) vs += __shfl_xor(vs, off, 32);
  if (lane == 0) red[wv] = vs;
  __syncthreads();
  float vtot = 0.f;
#pragma unroll
  for (int i = 0; i < 8; ++i) vtot += red[i];
  const float rstd = rsqrtf(vtot * (1.0f / D_) + 1e-5f);
#pragma unroll
  for (int i = 0; i < 4; ++i) {
    const int col = t + 256 * i;
    y[row * D_ + col] = (v[i] - mu) * rstd * g[col] + bta[col];
  }
}

// ---------------------------------------------------------------------------
extern "C" void kernel_launch(void* const* d_in, const int* in_sizes, int n_in,
                              void* d_out, int out_size, void* d_ws, size_t ws_size,
                              hipStream_t stream) {
  (void)in_sizes; (void)n_in; (void)out_size; (void)ws_size;
  const float* x   = (const float*)d_in[0];
  const float* Wq  = (const float*)d_in[1];
  const float* bq  = (const float*)d_in[2];
  const float* Wk  = (const float*)d_in[3];
  const float* bk  = (const float*)d_in[4];
  const float* Wv  = (const float*)d_in[5];
  const float* bv  = (const float*)d_in[6];
  const float* Wo  = (const float*)d_in[7];
  const float* bo  = (const float*)d_in[8];
  const float* lng = (const float*)d_in[9];
  const float* lnb = (const float*)d_in[10];
  const float* hv  = (const float*)d_in[11];
  const int*   hm  = (const int*)d_in[12];
  const unsigned char* am = (const unsigned char*)d_in[13];

  const size_t TOK = (size_t)B_ * S_ * D_;   // 4194304
  float* yOut    = (float*)d_out;
  float* attnOut = yOut + TOK;

  float* Qbuf = (float*)d_ws;
  float* Kbuf = Qbuf + TOK;
  float* Vbuf = Kbuf + TOK;
  float* Ctx  = Vbuf + TOK;

  dim3 blk(256);
  dim3 gemmGrid(D_ / 64, (B_ * S_) / 64);    // (16, 64)

  gemm_bias_kernel<<<gemmGrid, blk, 0, stream>>>(x, Wq, bq, Qbuf, B_ * S_, D_, D_);
  gemm_bias_kernel<<<gemmGrid, blk, 0, stream>>>(x, Wk, bk, Kbuf, B_ * S_, D_, D_);
  gemm_bias_kernel<<<gemmGrid, blk, 0, stream>>>(x, Wv, bv, Vbuf, B_ * S_, D_, D_);

  dim3 attGrid(S_ / QROWS, H_, B_);          // (16, 16, 8)
  const size_t dynLds =
      (size_t)(QROWS * QS_STR + 64 * KT_STR + QROWS * SB_STR) * sizeof(float);
  attention_kernel<<<attGrid, blk, dynLds, stream>>>(Qbuf, Kbuf, Vbuf, hv, hm, am,
                                                     attnOut, Ctx);

  // output projection (reuse Qbuf as scratch), then residual + LayerNorm
  gemm_bias_kernel<<<gemmGrid, blk, 0, stream>>>(Ctx, Wo, bo, Qbuf, B_ * S_, D_, D_);
  resid_ln_kernel<<<dim3(B_ * S_), blk, 0, stream>>>(Qbuf, x, lng, lnb, yOut);
}